// get_model_82059645157666
// MI455X (gfx1250) — compile-verified
//
#include <hip/hip_runtime.h>

typedef float v2f __attribute__((ext_vector_type(2)));
typedef float v8f __attribute__((ext_vector_type(8)));

#define TPB 256
#define TPB_FPS 1024

// ---------------------------------------------------------------------------
// (B,3,N) -> (B,N,3)
__global__ __launch_bounds__(TPB) void transpose_kernel(
    const float* __restrict__ in, float* __restrict__ out, int nB, int n) {
  int tid = blockIdx.x * TPB + threadIdx.x;
  int total = nB * n * 3;
  if (tid >= total) return;
  int c = tid % 3;
  int r = tid / 3;
  int pt = r % n;
  int b = r / n;
  out[tid] = in[((size_t)b * 3 + c) * n + pt];
}

// ---------------------------------------------------------------------------
// Farthest point sampling: one 1024-thread block per batch (32 waves on one
// WGP), dist array resident in LDS. The npoint iterations are inherently
// serial (each depends on the previous argmax), so minimize per-iteration
// latency: 8 strided points per thread + LDS tree reduction.
__global__ __launch_bounds__(TPB_FPS) void fps_kernel(
    const float* __restrict__ xyz, int* __restrict__ out_idx, int n, int npoint) {
  __shared__ float dist[8192];
  __shared__ float redV[TPB_FPS];
  __shared__ int redI[TPB_FPS];
  int b = blockIdx.x;
  const float* X = xyz + (size_t)b * n * 3;
  int t = threadIdx.x;
  for (int j = t; j < n; j += TPB_FPS) dist[j] = 1e10f;
  __syncthreads();
  int farthest = 0;
  for (int i = 0; i < npoint; ++i) {
    if (t == 0) out_idx[b * npoint + i] = farthest;
    float cx = X[farthest * 3 + 0];
    float cy = X[farthest * 3 + 1];
    float cz = X[farthest * 3 + 2];
    float bestV = -1.0f;
    int bestI = 0;
    for (int j = t; j < n; j += TPB_FPS) {
      float dx = X[j * 3 + 0] - cx;
      float dy = X[j * 3 + 1] - cy;
      float dz = X[j * 3 + 2] - cz;
      float d = dx * dx + dy * dy + dz * dz;
      float dd = fminf(dist[j], d);
      dist[j] = dd;
      if (dd > bestV) { bestV = dd; bestI = j; }
    }
    redV[t] = bestV;
    redI[t] = bestI;
    __syncthreads();
    for (int s = TPB_FPS / 2; s > 0; s >>= 1) {
      if (t < s) {
        if (redV[t + s] > redV[t] ||
            (redV[t + s] == redV[t] && redI[t + s] < redI[t])) {
          redV[t] = redV[t + s];
          redI[t] = redI[t + s];
        }
      }
      __syncthreads();
    }
    farthest = redI[0];
    __syncthreads();
  }
}

// ---------------------------------------------------------------------------
__global__ __launch_bounds__(TPB) void gather_xyz_kernel(
    const float* __restrict__ xyz, const int* __restrict__ fidx,
    float* __restrict__ out, int nB, int n, int S) {
  int tid = blockIdx.x * TPB + threadIdx.x;
  if (tid >= nB * S * 3) return;
  int c = tid % 3;
  int r = tid / 3;
  int s = r % S;
  int b = r / S;
  out[tid] = xyz[((size_t)b * n + fidx[b * S + s]) * 3 + c];
}

// ---------------------------------------------------------------------------
// First K indices (ascending) with d^2 <= r^2; pad with first hit.
__global__ __launch_bounds__(TPB) void ball_query_kernel(
    const float* __restrict__ xyz, const float* __restrict__ new_xyz,
    int* __restrict__ idx, int nB, int n, int S, int K, float r2) {
  int tid = blockIdx.x * TPB + threadIdx.x;
  if (tid >= nB * S) return;
  int b = tid / S;
  const float* q = new_xyz + (size_t)tid * 3;
  const float* X = xyz + (size_t)b * n * 3;
  int* o = idx + (size_t)tid * K;
  int cnt = 0, first = 0;
  for (int j = 0; j < n && cnt < K; ++j) {
    float dx = X[j * 3 + 0] - q[0];
    float dy = X[j * 3 + 1] - q[1];
    float dz = X[j * 3 + 2] - q[2];
    if (dx * dx + dy * dy + dz * dz <= r2) {
      if (cnt == 0) first = j;
      o[cnt++] = j;
    }
  }
  for (; cnt < K; ++cnt) o[cnt] = first;
}

// ---------------------------------------------------------------------------
// grouped[b,s,k,:] = concat(xyz[idx]-new_xyz, pts[idx]); width = 3 + Cpts
__global__ __launch_bounds__(TPB) void group_kernel(
    const float* __restrict__ xyz, const float* __restrict__ new_xyz,
    const float* __restrict__ pts, const int* __restrict__ idx,
    float* __restrict__ out, int nB, int n, int S, int K, int Cpts) {
  int Cin = 3 + Cpts;
  long total = (long)nB * S * K * Cin;
  long tid = (long)blockIdx.x * TPB + threadIdx.x;
  if (tid >= total) return;
  int c = (int)(tid % Cin);
  long row = tid / Cin;
  int k = (int)(row % K);
  long bs = row / K;
  int s = (int)(bs % S);
  int b = (int)(bs / S);
  int j = idx[((size_t)(b * S + s)) * K + k];
  float v;
  if (c < 3) {
    v = xyz[((size_t)b * n + j) * 3 + c] - new_xyz[((size_t)(b * S + s)) * 3 + c];
  } else {
    v = pts[((size_t)b * n + j) * Cpts + (c - 3)];
  }
  out[tid] = v;
}

// ---------------------------------------------------------------------------
// Fold BN into per-output-channel scale/shift: y = gemm*s + ((b-mean)*s + beta)
__global__ __launch_bounds__(TPB) void fold_bn_kernel(
    const float* __restrict__ b, const float* __restrict__ beta,
    const float* __restrict__ gamma, const float* __restrict__ mean,
    const float* __restrict__ var, float* __restrict__ scale,
    float* __restrict__ shift, int O, int hasBN) {
  int o = blockIdx.x * TPB + threadIdx.x;
  if (o >= O) return;
  if (hasBN) {
    float s = gamma[o] * rsqrtf(var[o] + 1e-5f);
    scale[o] = s;
    shift[o] = (b[o] - mean[o]) * s + beta[o];
  } else {
    scale[o] = 1.0f;
    shift[o] = b[o];
  }
}

// ---------------------------------------------------------------------------
// Zero-pad a weight matrix to Opad rows (used for the 13-class head).
__global__ __launch_bounds__(TPB) void pad_w_kernel(
    const float* __restrict__ W, float* __restrict__ Wp, int O, int C,
    int Opad) {
  int tid = blockIdx.x * TPB + threadIdx.x;
  if (tid >= Opad * C) return;
  int row = tid / C;
  int col = tid - row * C;
  Wp[tid] = (row < O) ? W[row * C + col] : 0.0f;
}

// ---------------------------------------------------------------------------
// Y[M,O] = relu?( (X[M,C] x W[O,C]^T) * scale[o] + shift[o] )
// Full-fp32 WMMA (V_WMMA_F32_16X16X4_F32). Each wave computes a 16x32 output
// strip (two 16x16 tiles sharing the A fragment -> halves X traffic).
// REQUIRES: M % 16 == 0; W has at least ceil(O/16)*16 rows.
__global__ __launch_bounds__(TPB) void gemm_bn_act_wmma(
    const float* __restrict__ X, const float* __restrict__ W,
    const float* __restrict__ scale, const float* __restrict__ shift,
    float* __restrict__ Y, int M, int C, int O, int relu) {
  int wave = threadIdx.x >> 5;
  int lane = threadIdx.x & 31;
  int tilesO = (O + 15) >> 4;
  int tilesO2 = (tilesO + 1) >> 1;  // 32-wide strips
  int tilesM = M >> 4;
  long total = (long)tilesO2 * tilesM;
  long tile = (long)blockIdx.x * (TPB / 32) + wave;
  if (tile >= total) return;  // wave-uniform -> EXEC all-ones at WMMA
  int tm = (int)(tile / tilesO2);
  int to2 = (int)(tile % tilesO2);
  int mbase = tm << 4;
  int obase = to2 << 5;
  bool has2 = (to2 * 2 + 1) < tilesO;  // wave-uniform
  int half = lane >> 4;   // 0: K pair {k,k+1}; 1: K pair {k+2,k+3}
  int l15 = lane & 15;
  const float* xr = X + (size_t)(mbase + l15) * C;
  const float* wr0 = W + (size_t)(obase + l15) * C;
  // alias second tile onto first when absent: hot loop stays branchless
  const float* wr1 = has2 ? (W + (size_t)(obase + 16 + l15) * C) : wr0;
  int koff = half << 1;
  __builtin_prefetch(xr + koff, 0, 0);
  __builtin_prefetch(wr0 + koff, 0, 0);
  __builtin_prefetch(wr1 + koff, 0, 0);
  v8f acc0 = {};
  v8f acc1 = {};
  int k = 0;
  for (; k + 16 <= C; k += 16) {
    const float* xk = xr + k + koff;
    const float* pk = wr0 + k + koff;
    const float* qk = wr1 + k + koff;
#pragma unroll
    for (int u = 0; u < 4; ++u) {
      v2f a, b0, b1;
      a.x = xk[u * 4];      a.y = xk[u * 4 + 1];
      b0.x = pk[u * 4];     b0.y = pk[u * 4 + 1];
      b1.x = qk[u * 4];     b1.y = qk[u * 4 + 1];
      acc0 = __builtin_amdgcn_wmma_f32_16x16x4_f32(false, a, false, b0,
                                                   (short)0, acc0, false, false);
      acc1 = __builtin_amdgcn_wmma_f32_16x16x4_f32(false, a, false, b1,
                                                   (short)0, acc1, false, false);
    }
  }
  for (; k + 4 <= C; k += 4) {
    const float* xk = xr + k + koff;
    v2f a, b0, b1;
    a.x = xk[0];              a.y = xk[1];
    b0.x = wr0[k + koff];     b0.y = wr0[k + koff + 1];
    b1.x = wr1[k + koff];     b1.y = wr1[k + koff + 1];
    acc0 = __builtin_amdgcn_wmma_f32_16x16x4_f32(false, a, false, b0, (short)0,
                                                 acc0, false, false);
    acc1 = __builtin_amdgcn_wmma_f32_16x16x4_f32(false, a, false, b1, (short)0,
                                                 acc1, false, false);
  }
  if (k < C) {  // K remainder (C % 4 in {1,2,3}), zero-padded registers
    int kk = k + koff;
    v2f a = {0.0f, 0.0f}, b0 = {0.0f, 0.0f}, b1 = {0.0f, 0.0f};
    if (kk < C) { a.x = xr[kk]; b0.x = wr0[kk]; b1.x = wr1[kk]; }
    if (kk + 1 < C) { a.y = xr[kk + 1]; b0.y = wr0[kk + 1]; b1.y = wr1[kk + 1]; }
    acc0 = __builtin_amdgcn_wmma_f32_16x16x4_f32(false, a, false, b0, (short)0,
                                                 acc0, false, false);
    acc1 = __builtin_amdgcn_wmma_f32_16x16x4_f32(false, a, false, b1, (short)0,
                                                 acc1, false, false);
  }
  // C/D layout: lane<16 -> N=lane, vgpr r -> M=r; lane>=16 -> N=lane-16, M=r+8
  int col0 = obase + l15;
  if (col0 < O) {
    float sc = scale[col0];
    float sh = shift[col0];
#pragma unroll
    for (int r = 0; r < 8; ++r) {
      int row = mbase + r + (half << 3);
      float v = acc0[r] * sc + sh;
      if (relu) v = fmaxf(v, 0.0f);
      Y[(size_t)row * O + col0] = v;
    }
  }
  int col1 = obase + 16 + l15;
  if (has2 && col1 < O) {
    float sc = scale[col1];
    float sh = shift[col1];
#pragma unroll
    for (int r = 0; r < 8; ++r) {
      int row = mbase + r + (half << 3);
      float v = acc1[r] * sc + sh;
      if (relu) v = fmaxf(v, 0.0f);
      Y[(size_t)row * O + col1] = v;
    }
  }
}

// ---------------------------------------------------------------------------
__global__ __launch_bounds__(TPB) void maxpool_kernel(
    const float* __restrict__ H, float* __restrict__ out, int nB, int S, int K,
    int O) {
  long tid = (long)blockIdx.x * TPB + threadIdx.x;
  long total = (long)nB * S * O;
  if (tid >= total) return;
  int o = (int)(tid % O);
  long bs = tid / O;
  float m = -3.4e38f;
  const float* base = H + ((size_t)bs * K) * O + o;
  for (int k = 0; k < K; ++k) m = fmaxf(m, base[(size_t)k * O]);
  out[tid] = m;
}

// ---------------------------------------------------------------------------
// 3-NN inverse-distance interpolation + concat(points1, interp)
__global__ __launch_bounds__(TPB) void fp_interp_kernel(
    const float* __restrict__ xyz1, const float* __restrict__ xyz2,
    const float* __restrict__ pts1, const float* __restrict__ pts2,
    float* __restrict__ out, int nB, int n1, int n2, int C1, int C2) {
  int tid = blockIdx.x * TPB + threadIdx.x;
  if (tid >= nB * n1) return;
  int b = tid / n1;
  const float* p = xyz1 + (size_t)tid * 3;
  const float* X2 = xyz2 + (size_t)b * n2 * 3;
  float d0 = 1e30f, d1 = 1e30f, d2 = 1e30f;
  int i0 = 0, i1 = 0, i2 = 0;
  for (int j = 0; j < n2; ++j) {
    float dx = X2[j * 3 + 0] - p[0];
    float dy = X2[j * 3 + 1] - p[1];
    float dz = X2[j * 3 + 2] - p[2];
    float d = dx * dx + dy * dy + dz * dz;
    if (d < d0) {
      d2 = d1; i2 = i1; d1 = d0; i1 = i0; d0 = d; i0 = j;
    } else if (d < d1) {
      d2 = d1; i2 = i1; d1 = d; i1 = j;
    } else if (d < d2) {
      d2 = d; i2 = j;
    }
  }
  float w0 = 1.0f / (d0 + 1e-8f);
  float w1 = 1.0f / (d1 + 1e-8f);
  float w2 = 1.0f / (d2 + 1e-8f);
  float ws = w0 + w1 + w2;
  w0 /= ws; w1 /= ws; w2 /= ws;
  float* orow = out + (size_t)tid * (C1 + C2);
  if (pts1 != nullptr) {
    const float* p1 = pts1 + (size_t)tid * C1;
    for (int c = 0; c < C1; ++c) orow[c] = p1[c];
  }
  const float* q0 = pts2 + ((size_t)b * n2 + i0) * C2;
  const float* q1 = pts2 + ((size_t)b * n2 + i1) * C2;
  const float* q2 = pts2 + ((size_t)b * n2 + i2) * C2;
  for (int c = 0; c < C2; ++c)
    orow[C1 + c] = w0 * q0[c] + w1 * q1[c] + w2 * q2[c];
}

// ---------------------------------------------------------------------------
__global__ __launch_bounds__(TPB) void log_softmax_kernel(
    const float* __restrict__ logits, float* __restrict__ out, int rows, int C) {
  int r = blockIdx.x * TPB + threadIdx.x;
  if (r >= rows) return;
  const float* h = logits + (size_t)r * C;
  float m = -3.4e38f;
  for (int c = 0; c < C; ++c) m = fmaxf(m, h[c]);
  float s = 0.0f;
  for (int c = 0; c < C; ++c) s += __expf(h[c] - m);
  float lse = m + __logf(s);
  for (int c = 0; c < C; ++c) out[(size_t)r * C + c] = h[c] - lse;
}

// ---------------------------------------------------------------------------
struct BN {
  const float *W, *b, *beta, *gamma, *mean, *var;
};

extern "C" void kernel_launch(void* const* d_in, const int* in_sizes, int n_in,
                              void* d_out, int out_size, void* d_ws,
                              size_t ws_size, hipStream_t stream) {
  (void)out_size; (void)ws_size;
  const int B = 2, N = 8192;
  static const int NP[7] = {8192, 4096, 1024, 256, 64, 16, 4};
  static const int PC[7] = {3, 64, 128, 256, 512, 1024, 1024};
  static const int SAK[6] = {32, 32, 32, 32, 32, 16};
  static const float SAR[6] = {0.1f, 0.2f, 0.4f, 0.8f, 1.6f, 3.2f};
  static const int SACH[6][4] = {{6, 32, 32, 64},      {67, 64, 64, 128},
                                 {131, 128, 128, 256}, {259, 256, 256, 512},
                                 {515, 512, 512, 1024},{1027, 1024, 1024, 1024}};
  static const int FPCH[6][4] = {{2048, 512, 512, 0}, {1024, 512, 512, 0},
                                 {768, 256, 256, 0},  {384, 256, 256, 0},
                                 {320, 256, 128, 0},  {128, 128, 128, 128}};
  static const int FPNL[6] = {2, 2, 2, 2, 2, 3};
  static const int FPOFF[6] = {0, 12, 24, 36, 48, 60};  // leaf offsets of fp groups

  // Locate xyz input by its unique flat size; params are the remaining leaves
  // in jax sorted-key order: fp(78) | head0(6) head1(2) | sa(108).
  const float* xyz_in;
  int pstart;
  if (in_sizes[0] == B * 3 * N) {
    xyz_in = (const float*)d_in[0];
    pstart = 1;
  } else {
    xyz_in = (const float*)d_in[n_in - 1];
    pstart = 0;
  }
  const float* leaf[256];
  int nleaf = n_in - 1;
  for (int i = 0; i < nleaf && i < 256; ++i)
    leaf[i] = (const float*)d_in[pstart + i];

  auto getBN = [&](int off) -> BN {
    BN r;
    r.W = leaf[off + 0]; r.b = leaf[off + 1]; r.beta = leaf[off + 2];
    r.gamma = leaf[off + 3]; r.mean = leaf[off + 4]; r.var = leaf[off + 5];
    return r;
  };

  // ---------------- workspace carve ----------------
  float* wsf = (float*)d_ws;
  size_t off = 0;
  auto alloc = [&](size_t nf) -> float* {
    float* p = wsf + off;
    off += (nf + 63) & ~(size_t)63;
    return p;
  };
  float* lxyz[7];
  float* lpts[7];
  for (int l = 0; l < 7; ++l) lxyz[l] = alloc((size_t)B * NP[l] * 3);
  lpts[0] = lxyz[0];
  for (int l = 1; l < 7; ++l) lpts[l] = alloc((size_t)B * NP[l] * PC[l]);
  float* scrA = alloc(262144ull * 64);   // 16.78M floats
  float* scrB = alloc(262144ull * 32);   // 8.39M floats
  float* scrC = alloc(5242880ull);       // grouped/concat inputs
  float* featA = alloc(2101248ull);
  float* featB = alloc(2101248ull);
  float* scl = alloc(1024);
  float* shf = alloc(1024);
  float* w2pad = alloc(16 * 128);        // head classifier padded to 16 rows
  int* fpsIdx = (int*)alloc((size_t)B * 4096);
  int* grpIdx = (int*)alloc((size_t)B * 4096 * 32);

  auto grid1 = [](long total) {
    return dim3((unsigned)((total + TPB - 1) / TPB));
  };
  auto launch_gemm = [&](const float* X, const float* W, float* Y, int M,
                         int C, int O, int relu) {
    int tilesO = (O + 15) / 16;
    int tilesO2 = (tilesO + 1) / 2;
    long tiles = (long)(M / 16) * tilesO2;
    unsigned blocks = (unsigned)((tiles + (TPB / 32) - 1) / (TPB / 32));
    gemm_bn_act_wmma<<<blocks, TPB, 0, stream>>>(X, W, scl, shf, Y, M, C, O,
                                                 relu);
  };

  // ---------------- l_xyz[0] / l_pts[0] ----------------
  transpose_kernel<<<grid1((long)B * N * 3), TPB, 0, stream>>>(xyz_in, lxyz[0],
                                                               B, N);

  // ---------------- SA stages ----------------
  for (int l = 0; l < 6; ++l) {
    int n = NP[l], S = NP[l + 1], K = SAK[l];
    float r2 = SAR[l] * SAR[l];
    fps_kernel<<<B, TPB_FPS, 0, stream>>>(lxyz[l], fpsIdx, n, S);
    gather_xyz_kernel<<<grid1((long)B * S * 3), TPB, 0, stream>>>(
        lxyz[l], fpsIdx, lxyz[l + 1], B, n, S);
    ball_query_kernel<<<grid1((long)B * S), TPB, 0, stream>>>(
        lxyz[l], lxyz[l + 1], grpIdx, B, n, S, K, r2);
    long rows = (long)B * S * K;
    int Cin = SACH[l][0];
    group_kernel<<<grid1(rows * Cin), TPB, 0, stream>>>(
        lxyz[l], lxyz[l + 1], lpts[l], grpIdx, scrC, B, n, S, K, PC[l]);
    const float* g = scrC;
    int c = Cin;
    float* dsts[3] = {scrA, scrB, scrA};
    for (int t = 0; t < 3; ++t) {
      int O = SACH[l][t + 1];
      BN bn = getBN(86 + (l * 3 + t) * 6);
      fold_bn_kernel<<<grid1(O), TPB, 0, stream>>>(bn.b, bn.beta, bn.gamma,
                                                   bn.mean, bn.var, scl, shf,
                                                   O, 1);
      launch_gemm(g, bn.W, dsts[t], (int)rows, c, O, 1);
      g = dsts[t];
      c = O;
    }
    maxpool_kernel<<<grid1((long)B * S * c), TPB, 0, stream>>>(g, lpts[l + 1],
                                                               B, S, K, c);
  }

  // ---------------- FP stages ----------------
  const float* feat = lpts[6];
  int cf = 1024;
  for (int jj = 0; jj < 6; ++jj) {
    int j = 5 - jj;
    int n1 = NP[j], n2 = NP[j + 1];
    int C1 = (j == 0) ? 0 : PC[j];
    fp_interp_kernel<<<grid1((long)B * n1), TPB, 0, stream>>>(
        lxyz[j], lxyz[j + 1], (j == 0) ? (const float*)nullptr : lpts[j], feat,
        scrC, B, n1, n2, C1, cf);
    int rows = B * n1;
    int c = C1 + cf;
    const float* g = scrC;
    int NL = FPNL[jj];
    for (int t = 0; t < NL; ++t) {
      int O = FPCH[jj][t + 1];
      BN bn = getBN(FPOFF[jj] + t * 6);
      fold_bn_kernel<<<grid1(O), TPB, 0, stream>>>(bn.b, bn.beta, bn.gamma,
                                                   bn.mean, bn.var, scl, shf,
                                                   O, 1);
      float* dst = (t == NL - 1) ? ((jj & 1) ? featB : featA)
                                 : ((t & 1) ? scrB : scrA);
      launch_gemm(g, bn.W, dst, rows, c, O, 1);
      g = dst;
      c = O;
    }
    feat = g;
    cf = c;
  }

  // ---------------- head ----------------
  int rows = B * N;
  BN h0 = getBN(78);
  fold_bn_kernel<<<grid1(128), TPB, 0, stream>>>(h0.b, h0.beta, h0.gamma,
                                                 h0.mean, h0.var, scl, shf,
                                                 128, 1);
  launch_gemm(feat, h0.W, scrA, rows, 128, 128, 1);
  const float* W2 = leaf[84];
  const float* b2 = leaf[85];
  pad_w_kernel<<<grid1(16 * 128), TPB, 0, stream>>>(W2, w2pad, 13, 128, 16);
  fold_bn_kernel<<<grid1(13), TPB, 0, stream>>>(b2, b2, b2, b2, b2, scl, shf,
                                                13, 0);
  launch_gemm(scrA, w2pad, scrB, rows, 128, 13, 0);
  log_softmax_kernel<<<grid1(rows), TPB, 0, stream>>>(scrB, (float*)d_out,
                                                      rows, 13);
}